// MaxLinear_64433099374740
// MI455X (gfx1250) — compile-verified
//
#include <hip/hip_runtime.h>
#include <stdint.h>

// Tropical (max-plus) linear: out[b,o] = max_i (W[o,i] + x[b,i])
// B=512, IN=1024, OUT=512, fp32.
// VALU-bound kernel (WMMA implements only the (*,+) semiring, not (max,+)).
// CDNA5 paths: global_load_async_to_lds_b128 (ASYNCcnt) double-buffered staging,
// s_wait_asynccnt split-counter waits, wave32 tiling, v_max3-friendly reduction.

#define B_    512
#define IN_   1024
#define OUT_  512

#define BM    64        // batch rows per workgroup
#define BN    32        // out cols per workgroup
#define BK    64        // K slab
#define STRIDE 68       // BK + 4 dword pad (breaks LDS bank conflicts; %4==0 for b128)
#define NTHREADS 256

// Builtin signature (confirmed by hipcc diagnostic in a previous round):
//   void __builtin_amdgcn_global_load_async_to_lds_b128(
//       v4i addrspace(1)*, v4i addrspace(3)*, imm int offset, imm int cpol)
typedef int v4i __attribute__((vector_size(4 * sizeof(int))));
typedef __attribute__((address_space(1))) v4i gv4i_t;
typedef __attribute__((address_space(3))) v4i lv4i_t;

__device__ __forceinline__ void async_copy_b128(const float* g, float* l) {
#if __has_builtin(__builtin_amdgcn_global_load_async_to_lds_b128)
  __builtin_amdgcn_global_load_async_to_lds_b128(
      (gv4i_t*)g, (lv4i_t*)l, /*offset=*/0, /*cpol=*/0);
#else
  *(float4*)l = *(const float4*)g;   // synchronous fallback
#endif
}

// Always emit a wait: builtin when available, raw CDNA5 mnemonic otherwise.
__device__ __forceinline__ void wait_async_le6(void) {
#if __has_builtin(__builtin_amdgcn_s_wait_asynccnt)
  __builtin_amdgcn_s_wait_asynccnt(6);
#else
  asm volatile("s_wait_asynccnt 6" ::: "memory");
#endif
}
__device__ __forceinline__ void wait_async_le0(void) {
#if __has_builtin(__builtin_amdgcn_s_wait_asynccnt)
  __builtin_amdgcn_s_wait_asynccnt(0);
#else
  asm volatile("s_wait_asynccnt 0" ::: "memory");
#endif
}

// Stage one BMxBK x-tile and BNxBK W-tile into LDS (6 async b128 per thread).
// xg/wg already point at this thread's chunk for the current K slab.
__device__ __forceinline__ void stage_tile(const float* xg, const float* wg,
                                           float* xsbuf, float* wsbuf,
                                           int xrow, int xkc, int wrow, int wkc) {
#pragma unroll
  for (int i = 0; i < 4; ++i) {
    // x tile: 64 rows x 16 b128-chunks; consecutive i steps 16 rows down.
    async_copy_b128(xg + (size_t)i * 16 * IN_,
                    &xsbuf[(xrow + i * 16) * STRIDE + xkc]);
  }
#pragma unroll
  for (int i = 0; i < 2; ++i) {
    // W tile: 32 rows x 16 b128-chunks; consecutive i steps 16 rows down.
    async_copy_b128(wg + (size_t)i * 16 * IN_,
                    &wsbuf[(wrow + i * 16) * STRIDE + wkc]);
  }
}

__global__ __launch_bounds__(NTHREADS)
void maxplus_linear_kernel(const float* __restrict__ x,
                           const float* __restrict__ W,
                           float* __restrict__ out) {
  __shared__ __align__(16) float xs[2][BM * STRIDE];  // 2 x 17408 B
  __shared__ __align__(16) float ws[2][BN * STRIDE];  // 2 x  8704 B

  const int tid = threadIdx.x;
  const int tx  = tid & 15;        // 16 col groups -> n = tx*2
  const int ty  = tid >> 4;        // 16 row groups -> m = ty*4
  const int m0  = blockIdx.y * BM;
  const int n0  = blockIdx.x * BN;

  // Per-thread staging coordinates (chunk id = tid, 16 chunks per row).
  const int srow = tid >> 4;            // 0..15
  const int skc  = (tid & 15) << 2;     // dword offset within row, 0..60
  // Incrementing global base pointers (advance by BK each K slab).
  const float* xg = x + (size_t)(m0 + srow) * IN_ + skc;
  const float* wg = W + (size_t)(n0 + srow) * IN_ + skc;

  float acc[4][2];
#pragma unroll
  for (int r = 0; r < 4; ++r)
#pragma unroll
    for (int c = 0; c < 2; ++c)
      acc[r][c] = -__builtin_inff();

  // Prologue: stage tile 0 into buffer 0.
  stage_tile(xg, wg, xs[0], ws[0], srow, skc, srow, skc);

  const int ntiles = IN_ / BK;     // 16
  for (int t = 0; t < ntiles; ++t) {
    const int cur = t & 1;

    // Overlap: stage tile t+1 into the other buffer, then wait for tile t.
    if (t + 1 < ntiles) {
      stage_tile(xg + (size_t)(t + 1) * BK, wg + (size_t)(t + 1) * BK,
                 xs[cur ^ 1], ws[cur ^ 1], srow, skc, srow, skc);
      wait_async_le6();            // ASYNCcnt <= 6: tile t's 6 loads done (in-order)
    } else {
      wait_async_le0();            // ASYNCcnt == 0
    }
    __syncthreads();               // all waves' staged data visible

    const float* xb = xs[cur];
    const float* wb = ws[cur];

#pragma unroll
    for (int kk = 0; kk < BK; kk += 4) {
      float4 xr[4], wr[2];
#pragma unroll
      for (int r = 0; r < 4; ++r)
        xr[r] = *(const float4*)&xb[(ty * 4 + r) * STRIDE + kk];
#pragma unroll
      for (int c = 0; c < 2; ++c)
        wr[c] = *(const float4*)&wb[(tx * 2 + c) * STRIDE + kk];

#pragma unroll
      for (int r = 0; r < 4; ++r) {
#pragma unroll
        for (int c = 0; c < 2; ++c) {
          const float s0 = xr[r].x + wr[c].x;
          const float s1 = xr[r].y + wr[c].y;
          const float s2 = xr[r].z + wr[c].z;
          const float s3 = xr[r].w + wr[c].w;
          // v_max3-friendly tree; single dep on acc per 4-k step
          acc[r][c] = fmaxf(acc[r][c], fmaxf(fmaxf(s0, s1), fmaxf(s2, s3)));
        }
      }
    }
    __syncthreads();               // protect buffer reuse next iteration
  }

  // Write 4x2 outputs per thread; n pair is contiguous -> b64 stores.
  const int m = m0 + ty * 4;
  const int n = n0 + tx * 2;
#pragma unroll
  for (int r = 0; r < 4; ++r) {
    float2 v;
    v.x = acc[r][0];
    v.y = acc[r][1];
    *(float2*)&out[(size_t)(m + r) * OUT_ + n] = v;
  }
}

extern "C" void kernel_launch(void* const* d_in, const int* in_sizes, int n_in,
                              void* d_out, int out_size, void* d_ws, size_t ws_size,
                              hipStream_t stream) {
  (void)in_sizes; (void)n_in; (void)out_size; (void)d_ws; (void)ws_size;
  const float* x = (const float*)d_in[0];   // [B, IN]
  const float* W = (const float*)d_in[1];   // [OUT, IN]
  float* out = (float*)d_out;               // [B, OUT]

  dim3 grid(OUT_ / BN, B_ / BM);            // 16 x 8 = 128 workgroups
  dim3 block(NTHREADS);
  hipLaunchKernelGGL(maxplus_linear_kernel, grid, block, 0, stream, x, W, out);
}